// QKVAttentionMapping_40209483825889
// MI455X (gfx1250) — compile-verified
//
#include <hip/hip_runtime.h>
#include <hip/hip_bf16.h>

typedef float v2f __attribute__((ext_vector_type(2)));
typedef float v8f __attribute__((ext_vector_type(8)));

#define S_LEN 1024
#define DHEAD 64
#define HALF_D 32
#define BT 128            // block tile (M and N)
#define LDSS 68           // padded LDS row stride (floats): 68*4=272 bytes, 16B aligned rows

// ---------------------------------------------------------------------------
// Kernel 1: interleaved-pair RoPE on Q and K, plus V passthrough copy.
// One thread per (b,h,s,i) pair; layout [B,H,S,d] => pair p has i=p%32,
// s=(p/32)%1024. Writes directly into the q/k/v output slots of d_out.
// ---------------------------------------------------------------------------
__global__ __launch_bounds__(256) void rope_pack_kernel(
    const float* __restrict__ q, const float* __restrict__ k,
    const float* __restrict__ v, float* __restrict__ oq,
    float* __restrict__ ok, float* __restrict__ ov, int npairs) {
  int p = blockIdx.x * blockDim.x + threadIdx.x;
  if (p >= npairs) return;
  int i = p & (HALF_D - 1);
  int s = (p >> 5) & (S_LEN - 1);
  // inv_freq = theta^(-2i/d) = exp(-i * ln(theta)/(d/2))
  float inv = __expf(-(float)i * (__logf(10000.0f) / (float)HALF_D));
  float ang = (float)s * inv;
  float sn, cs;
  __sincosf(ang, &sn, &cs);
  float2 xq = ((const float2*)q)[p];
  float2 xk = ((const float2*)k)[p];
  float2 rq = make_float2(xq.x * cs - xq.y * sn, xq.y * cs + xq.x * sn);
  float2 rk = make_float2(xk.x * cs - xk.y * sn, xk.y * cs + xk.x * sn);
  ((float2*)oq)[p] = rq;
  ((float2*)ok)[p] = rk;
  ((float2*)ov)[p] = ((const float2*)v)[p];
}

// ---------------------------------------------------------------------------
// Kernel 2: scores[bh] = Qr[bh] * Kr[bh]^T using V_WMMA_F32_16X16X4_F32.
// Block = 256 threads (8 wave32 waves) computing a 128x128 output tile.
// Wave w: mOff = (w>>1)*32, nOff = (w&1)*64 -> 2x4 grid of 16x16 tiles.
//
// ISA fragment layouts (32-bit, wave32):
//  A 16x4 : lane L holds row M=L&15, cols {k0+2*(L>>4), k0+2*(L>>4)+1}
//  B 4x16 : lane L holds col N=L&15, rows {k0+2*(L>>4), k0+2*(L>>4)+1}
//           and B[k][n] = Krope[n][k], so both fragments are the same
//           2-consecutive-dword LDS read pattern.
//  C/D    : VGPR r, lane L -> (m = r + 8*(L>>4), n = L&15)
// ---------------------------------------------------------------------------
__global__ __launch_bounds__(256) void attn_qk_wmma_kernel(
    const float* __restrict__ q, const float* __restrict__ k,
    float* __restrict__ aw) {
  __shared__ float ldsA[BT * LDSS];
  __shared__ float ldsB[BT * LDSS];

  const int bh = blockIdx.y;
  const int tm = blockIdx.x & 7;   // 1024/128 = 8 tiles per dim
  const int tn = blockIdx.x >> 3;
  const int t  = threadIdx.x;

  const float* qg = q + ((size_t)bh * S_LEN + (size_t)tm * BT) * DHEAD;
  const float* kg = k + ((size_t)bh * S_LEN + (size_t)tn * BT) * DHEAD;

  // Cooperative stage: 128 rows x 64 floats per matrix, float4 per thread,
  // 16 threads per row, 16 rows per pass, 8 passes.
  {
    const int col4 = (t & 15) * 4;
    const int rb   = t >> 4;
#pragma unroll
    for (int pr = 0; pr < 8; ++pr) {
      const int row = pr * 16 + rb;
      float4 aq = *(const float4*)(qg + row * DHEAD + col4);
      float4 ak = *(const float4*)(kg + row * DHEAD + col4);
      *(float4*)(&ldsA[row * LDSS + col4]) = aq;
      *(float4*)(&ldsB[row * LDSS + col4]) = ak;
    }
  }
  __syncthreads();

  const int wid  = t >> 5;
  const int lane = t & 31;
  const int lo   = lane & 15;
  const int hi   = lane >> 4;
  const int mOff = (wid >> 1) * 32;  // 0,32,64,96
  const int nOff = (wid & 1) * 64;   // 0,64

  v8f acc[2][4];
#pragma unroll
  for (int mt = 0; mt < 2; ++mt)
#pragma unroll
    for (int nt = 0; nt < 4; ++nt) acc[mt][nt] = {};

  for (int k0 = 0; k0 < DHEAD; k0 += 4) {
    const int col = k0 + 2 * hi;
    v2f a[2], b[4];
#pragma unroll
    for (int mt = 0; mt < 2; ++mt)
      a[mt] = *(const v2f*)(&ldsA[(mOff + mt * 16 + lo) * LDSS + col]);
#pragma unroll
    for (int nt = 0; nt < 4; ++nt)
      b[nt] = *(const v2f*)(&ldsB[(nOff + nt * 16 + lo) * LDSS + col]);
#pragma unroll
    for (int mt = 0; mt < 2; ++mt)
#pragma unroll
      for (int nt = 0; nt < 4; ++nt)
        acc[mt][nt] = __builtin_amdgcn_wmma_f32_16x16x4_f32(
            false, a[mt], false, b[nt], (short)0, acc[mt][nt], false, false);
  }

  // Store D tiles: row = tile_m + r + 8*hi, col = tile_n + lo
  float* o = aw + (size_t)bh * S_LEN * S_LEN;
#pragma unroll
  for (int mt = 0; mt < 2; ++mt) {
#pragma unroll
    for (int nt = 0; nt < 4; ++nt) {
      const int colo  = tn * BT + nOff + nt * 16 + lo;
      const int rbase = tm * BT + mOff + mt * 16 + 8 * hi;
#pragma unroll
      for (int r = 0; r < 8; ++r)
        o[(size_t)(rbase + r) * S_LEN + colo] = acc[mt][nt][r];
    }
  }
}

extern "C" void kernel_launch(void* const* d_in, const int* in_sizes, int n_in,
                              void* d_out, int out_size, void* d_ws, size_t ws_size,
                              hipStream_t stream) {
  // Input order: n_heads, d_queries, d_values, queries, keys, values,
  //              query_sequence_pad_length, key_value_sequence_pad_length
  const float* q = (const float*)d_in[3];
  const float* k = (const float*)d_in[4];
  const float* v = (const float*)d_in[5];

  const int q_elems = in_sizes[3];                 // B*H*S*d
  const int BH      = q_elems / (S_LEN * DHEAD);   // 128

  float* out_aw = (float*)d_out;
  float* out_q  = out_aw + (size_t)BH * S_LEN * S_LEN;
  float* out_k  = out_q + (size_t)q_elems;
  float* out_v  = out_k + (size_t)q_elems;

  // 1) RoPE(Q), RoPE(K), copy V into output slots.
  const int npairs = q_elems / 2;
  rope_pack_kernel<<<(npairs + 255) / 256, 256, 0, stream>>>(
      q, k, v, out_q, out_k, out_v, npairs);

  // 2) scores = Qr @ Kr^T per batch-head, WMMA f32 16x16x4.
  dim3 grid(64 /* 8x8 tiles */, BH);
  attn_qk_wmma_kernel<<<grid, 256, 0, stream>>>(out_q, out_k, out_aw);
}